// Inverse_ICP_20744692040054
// MI455X (gfx1250) — compile-verified
//
#include <hip/hip_runtime.h>
#include <math.h>

typedef __attribute__((ext_vector_type(2))) float v2f;
typedef __attribute__((ext_vector_type(8))) float v8f;

#define H_ 480
#define W_ 640
#define HW_ (H_ * W_)

__device__ __forceinline__ float fast_rcp(float x) { return __builtin_amdgcn_rcpf(x); }
__device__ __forceinline__ float fast_sqrt(float x) { return __builtin_amdgcn_sqrtf(x); }

// order-preserving float<->uint encoding for atomic min/max
__device__ __forceinline__ unsigned fenc(float f) {
  unsigned u = __float_as_uint(f);
  return (u & 0x80000000u) ? ~u : (u | 0x80000000u);
}
__device__ __forceinline__ float fdec(unsigned u) {
  return __uint_as_float((u & 0x80000000u) ? (u & 0x7FFFFFFFu) : ~u);
}

// ---------------------------------------------------------------- init
__global__ void k_init(const float* __restrict__ Rin, const float* __restrict__ tin,
                       float* __restrict__ Rt, float* __restrict__ gram,
                       unsigned* __restrict__ mm, int B) {
  int i = blockIdx.x * blockDim.x + threadIdx.x;
  if (i < B * 256) gram[i] = 0.f;
  if (i < B * 12) {
    int b = i / 12, j = i - b * 12;
    Rt[i] = (j < 9) ? Rin[b * 9 + j] : tin[b * 3 + (j - 9)];
  }
  if (i == 0) { mm[0] = 0xFFFFFFFFu; mm[1] = 0u; }
}

// ---------------------------------------------------------------- global min/max of depth1
__global__ void k_minmax(const float* __restrict__ d1, unsigned* __restrict__ mm, int total) {
  __shared__ unsigned smin, smax;
  if (threadIdx.x == 0) { smin = 0xFFFFFFFFu; smax = 0u; }
  __syncthreads();
  unsigned lmin = 0xFFFFFFFFu, lmax = 0u;
  for (int i = blockIdx.x * blockDim.x + threadIdx.x; i < total; i += gridDim.x * blockDim.x) {
    unsigned e = fenc(d1[i]);
    lmin = min(lmin, e);
    lmax = max(lmax, e);
  }
  atomicMin(&smin, lmin);
  atomicMax(&smax, lmax);
  __syncthreads();
  if (threadIdx.x == 0) { atomicMin(&mm[0], smin); atomicMax(&mm[1], smax); }
}

// ---------------------------------------------------------------- normal1 (Sobel on vertex1 from depth1), packed float4
__global__ __launch_bounds__(256) void k_normal(const float* __restrict__ depth1,
                                                const float* __restrict__ Km,
                                                const unsigned* __restrict__ mm,
                                                float4* __restrict__ nrm) {
  int b = blockIdx.y;
  const float fx = Km[b * 4 + 0], fy = Km[b * 4 + 1], cx = Km[b * 4 + 2], cy = Km[b * 4 + 3];
  const float ifx = 1.f / fx, ify = 1.f / fy; // hoisted: pixel loop is multiply-only
  const float* d = depth1 + (size_t)b * HW_;
  float4* nb = nrm + (size_t)b * HW_;
  float gmin = fdec(mm[0]), gmax = fdec(mm[1]);

  for (int n = blockIdx.x * blockDim.x + threadIdx.x; n < HW_; n += gridDim.x * blockDim.x) {
    int y = n / W_, x = n - y * W_;
    int ym = max(y - 1, 0), yp = min(y + 1, H_ - 1);
    int xm = max(x - 1, 0), xp = min(x + 1, W_ - 1);

#define VTX(yy, xx, vx, vy, vz)                           \
    {                                                     \
      float dd = d[(yy) * W_ + (xx)];                     \
      vx = ((float)(xx) - cx) * ifx * dd;                 \
      vy = ((float)(yy) - cy) * ify * dd;                 \
      vz = dd;                                            \
    }
    float ax, ay, az, bx, by, bz;
    float dxx = 0.f, dxy = 0.f, dxz = 0.f, dyx, dyy, dyz;

    VTX(ym, xp, ax, ay, az); VTX(ym, xm, bx, by, bz);
    dxx += ax - bx; dxy += ay - by; dxz += az - bz;
    float c_ym_xm_x = bx, c_ym_xm_y = by, c_ym_xm_z = bz;
    float c_ym_xp_x = ax, c_ym_xp_y = ay, c_ym_xp_z = az;
    VTX(y, xp, ax, ay, az); VTX(y, xm, bx, by, bz);
    dxx += 2.f * (ax - bx); dxy += 2.f * (ay - by); dxz += 2.f * (az - bz);
    VTX(yp, xp, ax, ay, az); VTX(yp, xm, bx, by, bz);
    dxx += ax - bx; dxy += ay - by; dxz += az - bz;
    float c_yp_xp_x = ax, c_yp_xp_y = ay, c_yp_xp_z = az;
    float c_yp_xm_x = bx, c_yp_xm_y = by, c_yp_xm_z = bz;
    dyx = c_yp_xm_x - c_ym_xm_x; dyy = c_yp_xm_y - c_ym_xm_y; dyz = c_yp_xm_z - c_ym_xm_z;
    VTX(yp, x, ax, ay, az); VTX(ym, x, bx, by, bz);
    dyx += 2.f * (ax - bx); dyy += 2.f * (ay - by); dyz += 2.f * (az - bz);
    dyx += c_yp_xp_x - c_ym_xp_x; dyy += c_yp_xp_y - c_ym_xp_y; dyz += c_yp_xp_z - c_ym_xp_z;
#undef VTX

    float nx = dxy * dyz - dxz * dyy;
    float ny = dxz * dyx - dxx * dyz;
    float nz = dxx * dyy - dxy * dyx;
    float mag = fast_sqrt(nx * nx + ny * ny + nz * nz + 1e-16f);
    float inv = fast_rcp(mag + 1e-8f);
    nx *= inv; ny *= inv; nz *= inv;
    float dc = d[n];
    if (dc == gmin || dc == gmax) { nx = 0.f; ny = 0.f; nz = 0.f; }
    nb[n] = make_float4(nx, ny, nz, 0.f); // one b128 store
  }
}

// ---------------------------------------------------------------- per-iteration: J rows + WMMA Gram accumulation
__global__ __launch_bounds__(256) void k_accum(const float* __restrict__ depth0,
                                               const float* __restrict__ depth1,
                                               const float* __restrict__ Km,
                                               const float4* __restrict__ nrm,
                                               const float* __restrict__ Rt,
                                               float* __restrict__ gram) {
  int b = blockIdx.y;
  const float fx = Km[b * 4 + 0], fy = Km[b * 4 + 1], cx = Km[b * 4 + 2], cy = Km[b * 4 + 3];
  const float ifx = 1.f / fx, ify = 1.f / fy; // hoisted reciprocals
  const float* rt = Rt + b * 12;
  float R00 = rt[0], R01 = rt[1], R02 = rt[2];
  float R10 = rt[3], R11 = rt[4], R12 = rt[5];
  float R20 = rt[6], R21 = rt[7], R22 = rt[8];
  float t0 = rt[9], t1 = rt[10], t2 = rt[11];

  const float* d0b = depth0 + (size_t)b * HW_;
  const float* d1b = depth1 + (size_t)b * HW_;
  const float4* nb = nrm + (size_t)b * HW_;

  __shared__ float smat[8][32][16]; // per-wave staging: 32 augmented rows x 16 (7 used)
  __shared__ float sgram[256];

  int lane = threadIdx.x & 31;
  int wave = threadIdx.x >> 5;
  for (int i = threadIdx.x; i < 256; i += blockDim.x) sgram[i] = 0.f;
  for (int j = 7; j < 16; ++j) smat[wave][lane][j] = 0.f; // static zero padding
  __syncthreads();

  v8f acc = {};
  int stride = gridDim.x * blockDim.x;
  int base = blockIdx.x * blockDim.x + threadIdx.x;
  int iters = (HW_ + stride - 1) / stride; // uniform trip count: EXEC all-ones at WMMA

  for (int it = 0; it < iters; ++it) {
    int n = base + it * stride;
    float r0 = 0.f, r1 = 0.f, r2 = 0.f, r3 = 0.f, r4 = 0.f, r5 = 0.f, r6 = 0.f;
    if (n < HW_) {
      int y = n / W_, x = n - y * W_;
      float d0 = d0b[n];
      float X0 = ((float)x - cx) * ifx * d0;
      float Y0 = ((float)y - cy) * ify * d0;
      float Z0 = d0;
      float xw = R00 * X0 + R01 * Y0 + R02 * Z0 + t0;
      float yw = R10 * X0 + R11 * Y0 + R12 * Z0 + t1;
      float zw = R20 * X0 + R21 * Y0 + R22 * Z0 + t2;
      float invz = fast_rcp(zw);
      float u_ = xw * invz * fx + cx;
      float v_ = yw * invz * fy + cy;
      bool inview = (u_ > 0.f) && (u_ < (float)(W_ - 1)) && (v_ > 0.f) && (v_ < (float)(H_ - 1));
      float uc = fminf(fmaxf(u_, 0.f), (float)(W_ - 1));
      float vc = fminf(fmaxf(v_, 0.f), (float)(H_ - 1));
      float u0f = floorf(uc), v0f = floorf(vc);
      float wu = uc - u0f, wv = vc - v0f;
      int u0 = (int)u0f, v0 = (int)v0f;
      int u1 = min(u0 + 1, W_ - 1), v1 = min(v0 + 1, H_ - 1);
      float w00 = (1.f - wu) * (1.f - wv), w10 = wu * (1.f - wv);
      float w01 = (1.f - wu) * wv, w11 = wu * wv;
      int i00 = v0 * W_ + u0, i10 = v0 * W_ + u1, i01 = v1 * W_ + u0, i11 = v1 * W_ + u1;
      float q00 = d1b[i00], q10 = d1b[i10], q01 = d1b[i01], q11 = d1b[i11];
      float pu0 = ((float)u0 - cx) * ifx, pu1 = ((float)u1 - cx) * ifx;
      float pv0 = ((float)v0 - cy) * ify, pv1 = ((float)v1 - cy) * ify;
      float rVx = w00 * pu0 * q00 + w10 * pu1 * q10 + w01 * pu0 * q01 + w11 * pu1 * q11;
      float rVy = w00 * pv0 * q00 + w10 * pv0 * q10 + w01 * pv1 * q01 + w11 * pv1 * q11;
      float rVz = w00 * q00 + w10 * q10 + w01 * q01 + w11 * q11;
      float4 n00 = nb[i00], n10 = nb[i10], n01 = nb[i01], n11 = nb[i11]; // 4x b128
      float rNx = w00 * n00.x + w10 * n10.x + w01 * n01.x + w11 * n11.x;
      float rNy = w00 * n00.y + w10 * n10.y + w01 * n01.y + w11 * n11.y;
      float rNz = w00 * n00.z + w10 * n10.z + w01 * n01.z + w11 * n11.z;
      float dxd = xw - rVx, dyd = yw - rVy, dzd = zw - rVz;
      float dnorm = fast_sqrt(dxd * dxd + dyd * dyd + dzd * dzd + 1e-16f);
      bool occ = (!inview) || (dnorm > 0.1f);
      if (!occ) {
        float res = rNx * dxd + rNy * dyd + rNz * dzd;
        float c0 = rNx * R00 + rNy * R10 + rNz * R20;
        float c1 = rNx * R01 + rNy * R11 + rNz * R21;
        float c2 = rNx * R02 + rNy * R12 + rNz * R22;
        float dz0 = d0 - 0.4f;
        float sigma = fabsf(c2) * (0.0012f + 0.0019f * dz0 * dz0) + 0.001f;
        float inv_s = fast_rcp(sigma + 1e-8f);
        // J = -[cross(NtC,P0), -NtC] / sigma ;  augmented 7th col = w_res
        r0 = -(c1 * Z0 - c2 * Y0) * inv_s;
        r1 = -(c2 * X0 - c0 * Z0) * inv_s;
        r2 = -(c0 * Y0 - c1 * X0) * inv_s;
        r3 = c0 * inv_s;
        r4 = c1 * inv_s;
        r5 = c2 * inv_s;
        r6 = res * inv_s;
      }
    }
    // stage this lane's augmented row (same-wave LDS in-order: store->load safe)
    float* row = &smat[wave][lane][0];
    row[0] = r0; row[1] = r1; row[2] = r2; row[3] = r3;
    row[4] = r4; row[5] = r5; row[6] = r6;
    __builtin_amdgcn_wave_barrier();

    // Gram += Rrows^T * Rrows via 8 x V_WMMA_F32_16X16X4_F32 (K=4 pixels each)
#pragma unroll
    for (int m = 0; m < 8; ++m) {
      int p0 = 4 * m + ((lane >> 4) << 1);
      int e = lane & 15;
      v2f a;
      a.x = smat[wave][p0][e];
      a.y = smat[wave][p0 + 1][e];
      acc = __builtin_amdgcn_wmma_f32_16x16x4_f32(false, a, false, a, (short)0, acc,
                                                  false, false);
    }
  }

  // C/D layout: VGPR g -> row (g + 8*(lane>>4)), col (lane&15)
  int mrow = (lane >> 4) * 8;
  int ncol = lane & 15;
#pragma unroll
  for (int g = 0; g < 8; ++g)
    atomicAdd(&sgram[(mrow + g) * 16 + ncol], acc[g]);
  __syncthreads();
  float* gb = gram + b * 256;
  for (int i = threadIdx.x; i < 256; i += blockDim.x)
    atomicAdd(&gb[i], sgram[i]);
}

// ---------------------------------------------------------------- per-batch 6x6 solve + exp-map update, re-zero gram
__global__ void k_solve(float* __restrict__ Rt, float* __restrict__ gram, int B) {
  int b = blockIdx.x;
  __shared__ float G[49];
  float* gb = gram + b * 256;
  for (int i = threadIdx.x; i < 49; i += blockDim.x)
    G[i] = gb[(i / 7) * 16 + (i % 7)];
  __syncthreads();
  for (int i = threadIdx.x; i < 256; i += blockDim.x) gb[i] = 0.f; // ready for next iteration
  if (threadIdx.x == 0) {
    float A[6][7];
    float tr = 0.f;
    for (int i = 0; i < 6; i++) tr += G[i * 7 + i];
    float lm = tr * 1e-6f;
    for (int i = 0; i < 6; i++) {
      for (int j = 0; j < 6; j++) A[i][j] = G[i * 7 + j] + (i == j ? lm : 0.f);
      A[i][6] = G[i * 7 + 6]; // JtR
    }
    // Gauss-Jordan with partial pivoting
    for (int k = 0; k < 6; k++) {
      int p = k;
      float best = fabsf(A[k][k]);
      for (int i = k + 1; i < 6; i++) {
        float v = fabsf(A[i][k]);
        if (v > best) { best = v; p = i; }
      }
      if (p != k)
        for (int j = 0; j < 7; j++) { float tmp = A[k][j]; A[k][j] = A[p][j]; A[p][j] = tmp; }
      float inv = 1.f / A[k][k];
      for (int j = 0; j < 7; j++) A[k][j] *= inv;
      for (int i = 0; i < 6; i++)
        if (i != k) {
          float f = A[i][k];
          for (int j = 0; j < 7; j++) A[i][j] -= f * A[k][j];
        }
    }
    float xi[6];
    for (int i = 0; i < 6; i++) xi[i] = A[i][6];
    // dR = twist2mat(-xi[0:3])
    float wx = -xi[0], wy = -xi[1], wz = -xi[2];
    float th = sqrtf(wx * wx + wy * wy + wz * wz + 1e-24f);
    float it = 1.f / th;
    float kx = wx * it, ky = wy * it, kz = wz * it;
    float s = sinf(th), c = 1.f - cosf(th);
    float dR[3][3];
    dR[0][0] = 1.f + c * (kx * kx - 1.f); dR[0][1] = -s * kz + c * kx * ky; dR[0][2] = s * ky + c * kx * kz;
    dR[1][0] = s * kz + c * kx * ky;      dR[1][1] = 1.f + c * (ky * ky - 1.f); dR[1][2] = -s * kx + c * ky * kz;
    dR[2][0] = -s * ky + c * kx * kz;     dR[2][1] = s * kx + c * ky * kz;  dR[2][2] = 1.f + c * (kz * kz - 1.f);
    float dt[3];
    for (int i = 0; i < 3; i++)
      dt[i] = -(dR[i][0] * xi[3] + dR[i][1] * xi[4] + dR[i][2] * xi[5]);
    float* rt = Rt + b * 12;
    float Ro[9], to[3], Rn[9], tn[3];
    for (int i = 0; i < 9; i++) Ro[i] = rt[i];
    for (int i = 0; i < 3; i++) to[i] = rt[9 + i];
    for (int i = 0; i < 3; i++) {
      for (int j = 0; j < 3; j++)
        Rn[i * 3 + j] = Ro[i * 3 + 0] * dR[0][j] + Ro[i * 3 + 1] * dR[1][j] + Ro[i * 3 + 2] * dR[2][j];
      tn[i] = Ro[i * 3 + 0] * dt[0] + Ro[i * 3 + 1] * dt[1] + Ro[i * 3 + 2] * dt[2] + to[i];
    }
    for (int i = 0; i < 9; i++) rt[i] = Rn[i];
    for (int i = 0; i < 3; i++) rt[9 + i] = tn[i];
  }
}

// ---------------------------------------------------------------- write outputs: R | t | weights(=1)
__global__ void k_out(const float* __restrict__ Rt, float* __restrict__ out, int B, int total) {
  int nR = B * 9, nT = B * 3;
  for (int i = blockIdx.x * blockDim.x + threadIdx.x; i < total; i += gridDim.x * blockDim.x) {
    float v;
    if (i < nR) {
      int b = i / 9;
      v = Rt[b * 12 + (i - b * 9)];
    } else if (i < nR + nT) {
      int j = i - nR;
      int b = j / 3;
      v = Rt[b * 12 + 9 + (j - b * 3)];
    } else {
      v = 1.0f;
    }
    out[i] = v;
  }
}

extern "C" void kernel_launch(void* const* d_in, const int* in_sizes, int n_in,
                              void* d_out, int out_size, void* d_ws, size_t ws_size,
                              hipStream_t stream) {
  const float* depth0 = (const float*)d_in[0];
  const float* depth1 = (const float*)d_in[1];
  const float* Km = (const float*)d_in[2];
  const float* Rin = (const float*)d_in[3];
  const float* tin = (const float*)d_in[4];
  float* out = (float*)d_out;

  int B = in_sizes[2] / 4; // K is (B,4)
  size_t HW = (size_t)H_ * W_;

  // workspace layout: normal1 float4[B*HW] | Rt[B*12] | gram[B*256] | minmax[2 uints]
  float4* nrm = (float4*)d_ws;
  float* Rt = (float*)d_ws + (size_t)4 * B * HW;
  float* gram = Rt + (size_t)B * 12;
  unsigned* mm = (unsigned*)(gram + (size_t)B * 256);

  int ninit = B * 256;
  k_init<<<dim3((ninit + 255) / 256), dim3(256), 0, stream>>>(Rin, tin, Rt, gram, mm, B);
  k_minmax<<<dim3(1024), dim3(256), 0, stream>>>(depth1, mm, B * (int)HW);
  k_normal<<<dim3(600, B), dim3(256), 0, stream>>>(depth1, Km, mm, nrm);
  for (int iter = 0; iter < 3; ++iter) {
    k_accum<<<dim3(120, B), dim3(256), 0, stream>>>(depth0, depth1, Km, nrm, Rt, gram);
    k_solve<<<dim3(B), dim3(64), 0, stream>>>(Rt, gram, B);
  }
  k_out<<<dim3(4096), dim3(256), 0, stream>>>(Rt, out, B, out_size);
}